// GRNN_62826781606047
// MI455X (gfx1250) — compile-verified
//
#include <hip/hip_runtime.h>
#include <hip/hip_bf16.h>

#define FD 64            // feature dim
#define WROW 68          // padded LDS row stride (floats): conflict-free b64 reads
#define NROWS_LDS 384    // 192 rows w_ih + 192 rows w_hh
#define LDS_BYTES (NROWS_LDS * WROW * 4)
#define ITERS 8          // row-block iterations per workgroup (256 rows/WG)

typedef __attribute__((ext_vector_type(2))) float v2f;
typedef __attribute__((ext_vector_type(8))) float v8f;

__device__ __forceinline__ v8f wmma_f32_4(v2f a, v2f b, v8f c) {
  // D = A(16x4) * B(4x16) + C, fp32 exact
  return __builtin_amdgcn_wmma_f32_16x16x4_f32(
      /*neg_a=*/false, a, /*neg_b=*/false, b,
      /*c_mod=*/(short)0, c, /*reuse_a=*/false, /*reuse_b=*/false);
}

__device__ __forceinline__ float fast_sigmoid(float x) {
  return 1.0f / (1.0f + __expf(-x));  // v_exp_f32 based
}

__device__ __forceinline__ float fast_tanh(float x) {
#if __has_builtin(__builtin_amdgcn_tanhf)
  return __builtin_amdgcn_tanhf(x);      // native gfx1250 v_tanh_f32
#elif __has_builtin(__builtin_amdgcn_tanh_f32)
  return __builtin_amdgcn_tanh_f32(x);
#else
  float e = __expf(2.0f * x);
  return 1.0f - 2.0f / (e + 1.0f);
#endif
}

// ---------------- Kernel 1: zero the aggregation buffer ----------------
__global__ void grnn_zero(float4* __restrict__ p, int n4) {
  int t = blockIdx.x * blockDim.x + threadIdx.x;
  if (t < n4) p[t] = make_float4(0.f, 0.f, 0.f, 0.f);
}

// ---------------- Kernel 2: scatter-add edges into receivers ----------------
// one thread per (edge, 4 features): float4 coalesced load + 4 fp32 atomics (L2-resident target)
__global__ void grnn_scatter(const float* __restrict__ edges,
                             const int* __restrict__ receivers,
                             float* __restrict__ aggr, int n_edges) {
  int t = blockIdx.x * blockDim.x + threadIdx.x;
  int total = n_edges * 16;
  if (t >= total) return;
  int e  = t >> 4;
  int d4 = (t & 15) * 4;
  const float4 v = *(const float4*)(edges + e * FD + d4);
  int r = receivers[e];
  float* p = aggr + r * FD + d4;
  (void)__hip_atomic_fetch_add(p + 0, v.x, __ATOMIC_RELAXED, __HIP_MEMORY_SCOPE_AGENT);
  (void)__hip_atomic_fetch_add(p + 1, v.y, __ATOMIC_RELAXED, __HIP_MEMORY_SCOPE_AGENT);
  (void)__hip_atomic_fetch_add(p + 2, v.z, __ATOMIC_RELAXED, __HIP_MEMORY_SCOPE_AGENT);
  (void)__hip_atomic_fetch_add(p + 3, v.w, __ATOMIC_RELAXED, __HIP_MEMORY_SCOPE_AGENT);
}

// ---------------- Kernel 3: fused dual-GEMM + GRU gates, LDS-staged weights ----------------
// block = 256 threads = 8 waves: wave = (row-sub-block rsub in {0,1}) x (col block c in 0..3).
// Both weight matrices staged once into 102KB of LDS (row stride 68 -> conflict-free ds_load_b64),
// then ITERS row-block iterations of 32 nodes reuse them (16x less weight re-read traffic).
__global__ __launch_bounds__(256) void grnn_gemm_gru(
    const float* __restrict__ aggr, const float* __restrict__ nodes,
    const float* __restrict__ w_ih, const float* __restrict__ w_hh,
    const float* __restrict__ b_ih, const float* __restrict__ b_hh,
    float* __restrict__ out, int n_nodes) {
  extern __shared__ float lds_w[];  // [384][68] : rows 0..191 = w_ih, 192..383 = w_hh

  // ---- cooperative weight stage: 384 rows x 16 float4 chunks ----
  for (int i = threadIdx.x; i < NROWS_LDS * (FD / 4); i += 256) {
    int row = i >> 4;
    int c4  = (i & 15) * 4;
    const float* src = (row < 192) ? (w_ih + row * FD + c4)
                                   : (w_hh + (row - 192) * FD + c4);
    *(float4*)&lds_w[row * WROW + c4] = *(const float4*)src;
  }
  __syncthreads();

  const int wave = threadIdx.x >> 5;
  const int cblk = wave & 3;           // feature column block 0..3
  const int rsub = wave >> 2;          // row sub-block 0..1
  const int lane = threadIdx.x & 31;
  const int nn   = lane & 15;          // N (B/C/D) and M (A rows) index
  const int kh   = lane >> 4;          // K-half select
  const int feat = cblk * 16 + nn;     // output feature column 0..63

  // bias scalars (splat into C: bias depends only on N)
  const float bir = b_ih[0 * FD + feat], biz = b_ih[1 * FD + feat], bin_ = b_ih[2 * FD + feat];
  const float bhr = b_hh[0 * FD + feat], bhz = b_hh[1 * FD + feat], bhn = b_hh[2 * FD + feat];

  // B fragments live in LDS: B[k][n] = w[gate*64+feat][k]
  const float* lwir = lds_w + (0 * FD + feat) * WROW + 2 * kh;
  const float* lwiz = lds_w + (1 * FD + feat) * WROW + 2 * kh;
  const float* lwin = lds_w + (2 * FD + feat) * WROW + 2 * kh;
  const float* lwhr = lds_w + (192 + 0 * FD + feat) * WROW + 2 * kh;
  const float* lwhz = lds_w + (192 + 1 * FD + feat) * WROW + 2 * kh;
  const float* lwhn = lds_w + (192 + 2 * FD + feat) * WROW + 2 * kh;

  int m_base = blockIdx.x * (32 * ITERS) + rsub * 16;
  for (int it = 0; it < ITERS; ++it, m_base += 32) {
    if (m_base >= n_nodes) break;  // wave-uniform: EXEC stays all-ones for WMMA

    v8f a_ir, a_iz, a_in, a_hr, a_hz, a_hn;
#pragma unroll
    for (int v = 0; v < 8; ++v) {
      a_ir[v] = bir; a_iz[v] = biz; a_in[v] = bin_;
      a_hr[v] = bhr; a_hz[v] = bhz; a_hn[v] = bhn;
    }

    // A fragment rows (clamped for the tail; no EXEC perturbation)
    const int mrow = min(m_base + nn, n_nodes - 1);
    const float* aRow = aggr  + mrow * FD + 2 * kh;
    const float* hRow = nodes + mrow * FD + 2 * kh;

#pragma unroll
    for (int k = 0; k < FD; k += 4) {
      v2f ai  = *(const v2f*)(aRow + k);
      v2f ah  = *(const v2f*)(hRow + k);
      v2f bir2 = *(const v2f*)(lwir + k);
      v2f biz2 = *(const v2f*)(lwiz + k);
      v2f bin2 = *(const v2f*)(lwin + k);
      v2f bhr2 = *(const v2f*)(lwhr + k);
      v2f bhz2 = *(const v2f*)(lwhz + k);
      v2f bhn2 = *(const v2f*)(lwhn + k);
      a_ir = wmma_f32_4(ai, bir2, a_ir);
      a_iz = wmma_f32_4(ai, biz2, a_iz);
      a_in = wmma_f32_4(ai, bin2, a_in);
      a_hr = wmma_f32_4(ah, bhr2, a_hr);
      a_hz = wmma_f32_4(ah, bhz2, a_hz);
      a_hn = wmma_f32_4(ah, bhn2, a_hn);
    }

    // GRU gate math on the 16x16 tile, fused store (divergence fine: WMMA done)
#pragma unroll
    for (int v = 0; v < 8; ++v) {
      int m = m_base + v + 8 * kh;
      if (m < n_nodes) {
        float h  = nodes[m * FD + feat];
        float r  = fast_sigmoid(a_ir[v] + a_hr[v]);
        float z  = fast_sigmoid(a_iz[v] + a_hz[v]);
        float nv = fast_tanh(a_in[v] + r * a_hn[v]);
        out[m * FD + feat] = (1.0f - z) * nv + z * h;
      }
    }
  }
}

extern "C" void kernel_launch(void* const* d_in, const int* in_sizes, int n_in,
                              void* d_out, int out_size, void* d_ws, size_t ws_size,
                              hipStream_t stream) {
  const float* edges     = (const float*)d_in[0];
  const float* nodes     = (const float*)d_in[1];
  const int*   receivers = (const int*)d_in[2];
  const float* w_ih      = (const float*)d_in[3];
  const float* w_hh      = (const float*)d_in[4];
  const float* b_ih      = (const float*)d_in[5];
  const float* b_hh      = (const float*)d_in[6];
  float* out = (float*)d_out;

  const int n_edges = in_sizes[2];
  const int n_nodes = in_sizes[1] / FD;

  float* aggr = (float*)d_ws;  // n_nodes * 64 floats (25.6 MB) in scratch

  // 1) zero aggr (every call: harness does not re-poison between replays)
  {
    int n4 = n_nodes * (FD / 4);
    int blocks = (n4 + 255) / 256;
    grnn_zero<<<blocks, 256, 0, stream>>>((float4*)aggr, n4);
  }
  // 2) scatter-add edges -> aggr
  {
    long long total = (long long)n_edges * 16;
    int blocks = (int)((total + 255) / 256);
    grnn_scatter<<<blocks, 256, 0, stream>>>(edges, receivers, aggr, n_edges);
  }
  // 3) fused GEMM + GRU with LDS-staged weights (256 rows per workgroup)
  {
    int rows_per_block = 32 * ITERS;
    int blocks = (n_nodes + rows_per_block - 1) / rows_per_block;
    grnn_gemm_gru<<<blocks, 256, LDS_BYTES, stream>>>(aggr, nodes, w_ih, w_hh,
                                                      b_ih, b_hh, out, n_nodes);
  }
}